// SymmetryLoss_86638080295595
// MI455X (gfx1250) — compile-verified
//
#include <hip/hip_runtime.h>
#include <math.h>

#define GS 32
#define GS3 (GS * GS * GS)          // 32768
// grid_min = -0.5 + 0.5/32 = -0.484375 ; scale = gs/(2*GB) = 32
#define GRID_OFF 0.484375f
#define GRID_SCALE 32.0f

typedef float v2f __attribute__((ext_vector_type(2)));
typedef float v8f __attribute__((ext_vector_type(8)));

// Masked squared distance for one transformed point.
// smask: 4KB LDS occupancy bitmask (bit=1 -> voxel occupied -> mask=0 -> no term)
__device__ __forceinline__ float dist2_term(float x, float y, float z, bool valid,
                                            const unsigned* smask, const float* cp_base) {
    if (!valid) return 0.0f;
    float fx = rintf(fminf(fmaxf((x + GRID_OFF) * GRID_SCALE, 0.0f), 31.0f));
    float fy = rintf(fminf(fmaxf((y + GRID_OFF) * GRID_SCALE, 0.0f), 31.0f));
    float fz = rintf(fminf(fmaxf((z + GRID_OFF) * GRID_SCALE, 0.0f), 31.0f));
    int ix = (int)fx, iy = (int)fy, iz = (int)fz;
    int flat = (ix << 10) | (iy << 5) | iz;
    if ((smask[flat >> 5] >> (flat & 31)) & 1u) return 0.0f;   // voxel occupied -> mask 0
    const float* c = cp_base + (size_t)flat * 3;
    float dx = x - c[0], dy = y - c[1], dz = z - c[2];
    return fmaf(dx, dx, fmaf(dy, dy, dz * dz));
}

// One block per batch. 256 threads = 8 wave32.
__global__ void __launch_bounds__(256)
symmetry_loss_main(const float* __restrict__ points,        // (B,N,3)
                   const float* __restrict__ control_points,// (B,32768,3)
                   const float* __restrict__ voxel,         // (B,32768)
                   const float* __restrict__ planes,        // (3,B,4)
                   const float* __restrict__ quats,         // (3,B,4)
                   float* __restrict__ ws,                  // [B] refl partials, [B..2B) rot partials
                   int Bsz, int N) {
    __shared__ unsigned smask[GS3 / 32];     // 4 KB occupancy bitmask
    __shared__ float strans[6][12];          // 6 affine transforms, row j = (M[j][0..2], t_j)
    __shared__ float sred[256];

    const int b   = blockIdx.x;
    const int tid = threadIdx.x;
    const int lane = tid & 31;
    const int col  = lane & 15;              // point column / A-matrix row index
    const bool hi  = lane >= 16;

    // ---- Phase 0a: pack voxel occupancy (32768 floats) into LDS bitmask ----
    {
        const float* vox = voxel + (size_t)b * GS3;
        for (int w = tid; w < GS3 / 32; w += 256) {
            const float4* v4 = reinterpret_cast<const float4*>(vox + (size_t)w * 32);
            unsigned m = 0;
            #pragma unroll
            for (int j = 0; j < 8; ++j) {
                float4 f = v4[j];
                m |= (f.x != 0.0f ? 1u : 0u) << (4 * j + 0);
                m |= (f.y != 0.0f ? 1u : 0u) << (4 * j + 1);
                m |= (f.z != 0.0f ? 1u : 0u) << (4 * j + 2);
                m |= (f.w != 0.0f ? 1u : 0u) << (4 * j + 3);
            }
            smask[w] = m;
        }
    }

    // ---- Phase 0b: build the 6 affine transforms (threads 0..5) ----
    if (tid < 6) {
        float m[12];
        if (tid < 3) {  // reflection: M = I - (2/n.n) n n^T ; t = -(2/n.n) d n
            const float* pl = planes + ((size_t)tid * Bsz + b) * 4;
            float a = pl[0], bb = pl[1], cc = pl[2], dd = pl[3];
            float k = 2.0f / (a * a + bb * bb + cc * cc);
            m[0] = 1.0f - k * a * a;  m[1] = -k * a * bb;        m[2] = -k * a * cc;        m[3]  = -k * dd * a;
            m[4] = -k * bb * a;       m[5] = 1.0f - k * bb * bb; m[6] = -k * bb * cc;       m[7]  = -k * dd * bb;
            m[8] = -k * cc * a;       m[9] = -k * cc * bb;       m[10] = 1.0f - k * cc * cc; m[11] = -k * dd * cc;
        } else {        // rotation from normalized quaternion (w,x,y,z); t = 0
            const float* qp = quats + ((size_t)(tid - 3) * Bsz + b) * 4;
            float w = qp[0], x = qp[1], y = qp[2], z = qp[3];
            float inv = 1.0f / sqrtf(w * w + x * x + y * y + z * z);
            w *= inv; x *= inv; y *= inv; z *= inv;
            m[0] = 1.0f - 2.0f * (y * y + z * z); m[1] = 2.0f * (x * y - w * z);        m[2] = 2.0f * (x * z + w * y);        m[3]  = 0.0f;
            m[4] = 2.0f * (x * y + w * z);        m[5] = 1.0f - 2.0f * (x * x + z * z); m[6] = 2.0f * (y * z - w * x);        m[7]  = 0.0f;
            m[8] = 2.0f * (x * z - w * y);        m[9] = 2.0f * (y * z + w * x);        m[10] = 1.0f - 2.0f * (x * x + y * y); m[11] = 0.0f;
        }
        #pragma unroll
        for (int j = 0; j < 12; ++j) strans[tid][j] = m[j];
    }
    __syncthreads();

    // ---- Phase 1: per-lane A-matrix registers (wave-invariant across the loop) ----
    // WMMA1 A rows: 0-2 = t0(refl), 3-5 = t1(refl), 8-10 = t2(refl), 11-13 = t3(rot)
    // WMMA2 A rows: 0-2 = t4(rot),  3-5 = t5(rot),  rest zero
    // A layout (16x4 f32): lane L<16 holds A[L][0..1]; lane L>=16 holds A[L-16][2..3]
    const int cA = hi ? 2 : 0;
    v2f A1v = {0.0f, 0.0f}, A2v = {0.0f, 0.0f};
    {
        int tr1 = -1, j1 = 0;
        if (col < 3)                  { tr1 = 0; j1 = col; }
        else if (col < 6)             { tr1 = 1; j1 = col - 3; }
        else if (col >= 8 && col < 11){ tr1 = 2; j1 = col - 8; }
        else if (col >= 11 && col < 14){ tr1 = 3; j1 = col - 11; }
        if (tr1 >= 0) { A1v.x = strans[tr1][j1 * 4 + cA]; A1v.y = strans[tr1][j1 * 4 + cA + 1]; }
        if (col < 3)      { A2v.x = strans[4][col * 4 + cA];       A2v.y = strans[4][col * 4 + cA + 1]; }
        else if (col < 6) { A2v.x = strans[5][(col - 3) * 4 + cA]; A2v.y = strans[5][(col - 3) * 4 + cA + 1]; }
    }

    // ---- Phase 2: main loop — 16 points per wave iteration via WMMA ----
    const float* pts = points + (size_t)b * N * 3;
    const float* cpb = control_points + (size_t)b * GS3 * 3;
    const int wid = tid >> 5;
    float accR = 0.0f, accT = 0.0f;

    for (int base = wid * 16; base < N; base += 128) {
        int pc = base + col;
        bool valid = (pc < N);
        // B layout (4x16 f32): lane c<16 -> (B[0][c],B[1][c]) = (x,y);
        //                      lane 16+c -> (B[2][c],B[3][c]) = (z,1)
        float b0 = 0.0f, b1 = 0.0f;
        if (valid) {
            const float* pp = pts + (size_t)pc * 3;
            if (!hi) { b0 = pp[0]; b1 = pp[1]; }
            else     { b0 = pp[2]; b1 = 1.0f; }
        }
        v2f Bv = {b0, b1};
        v8f z8 = {};
        // D = A x B (+0). EXEC is full here (guards only wrap loads/accumulates).
        v8f d1 = __builtin_amdgcn_wmma_f32_16x16x4_f32(false, A1v, false, Bv,
                                                       (short)0, z8, false, false);
        v8f d2 = __builtin_amdgcn_wmma_f32_16x16x4_f32(false, A2v, false, Bv,
                                                       (short)0, z8, false, false);
        // D rows 0-7 -> lanes 0-15 ; rows 8-15 -> lanes 16-31, column = lane&15.
        // slot0 of d1: t0 (lo) / t2 (hi) — both reflections
        accR += dist2_term(d1[0], d1[1], d1[2], valid, smask, cpb);
        // slot1 of d1: t1 (lo, refl) / t3 (hi, rot)
        {
            float s = dist2_term(d1[3], d1[4], d1[5], valid, smask, cpb);
            if (hi) accT += s; else accR += s;
        }
        // d2: lanes 0-15 carry t4, t5 (rotations); hi-lane rows are zero
        if (!hi) {
            accT += dist2_term(d2[0], d2[1], d2[2], valid, smask, cpb);
            accT += dist2_term(d2[3], d2[4], d2[5], valid, smask, cpb);
        }
    }

    // ---- Phase 3: deterministic block reduction -> workspace partials ----
    __syncthreads();
    sred[tid] = accR;
    __syncthreads();
    for (int s = 128; s > 0; s >>= 1) {
        if (tid < s) sred[tid] += sred[tid + s];
        __syncthreads();
    }
    if (tid == 0) ws[b] = sred[0];
    __syncthreads();
    sred[tid] = accT;
    __syncthreads();
    for (int s = 128; s > 0; s >>= 1) {
        if (tid < s) sred[tid] += sred[tid + s];
        __syncthreads();
    }
    if (tid == 0) ws[Bsz + b] = sred[0];
}

// Single-block, fixed-order final reduction: out[0]=refl, out[1]=rot (mean over B)
__global__ void __launch_bounds__(256)
symmetry_loss_reduce(const float* __restrict__ ws, float* __restrict__ out, int Bsz) {
    __shared__ float s0[256], s1[256];
    int t = threadIdx.x;
    float r0 = 0.0f, r1 = 0.0f;
    for (int i = t; i < Bsz; i += 256) {
        r0 += ws[i];
        r1 += ws[Bsz + i];
    }
    s0[t] = r0; s1[t] = r1;
    __syncthreads();
    for (int s = 128; s > 0; s >>= 1) {
        if (t < s) { s0[t] += s0[t + s]; s1[t] += s1[t + s]; }
        __syncthreads();
    }
    if (t == 0) {
        float invB = 1.0f / (float)Bsz;
        out[0] = s0[0] * invB;
        out[1] = s1[0] * invB;
    }
}

extern "C" void kernel_launch(void* const* d_in, const int* in_sizes, int n_in,
                              void* d_out, int out_size, void* d_ws, size_t ws_size,
                              hipStream_t stream) {
    const float* points         = (const float*)d_in[0]; // (B,N,3)
    const float* control_points = (const float*)d_in[1]; // (B,32768,3)
    const float* voxel          = (const float*)d_in[2]; // (B,1,32,32,32)
    const float* planes         = (const float*)d_in[3]; // (3,B,4)
    const float* quats          = (const float*)d_in[4]; // (3,B,4)
    float* out = (float*)d_out;
    float* ws  = (float*)d_ws;

    const int Bsz = in_sizes[3] / 12;            // planes = 3*B*4
    const int N   = in_sizes[0] / (3 * Bsz);     // points = B*N*3

    symmetry_loss_main<<<Bsz, 256, 0, stream>>>(points, control_points, voxel,
                                                planes, quats, ws, Bsz, N);
    symmetry_loss_reduce<<<1, 256, 0, stream>>>(ws, out, Bsz);
}